// ResidualVectorQuantizer_87703232184513
// MI455X (gfx1250) — compile-verified
//
#include <hip/hip_runtime.h>

typedef __attribute__((ext_vector_type(2))) float v2f;
typedef __attribute__((ext_vector_type(8))) float v8f;

#define N_TOK   32768          // 8*4096 rows
#define DIM     128
#define KCODES  1024
#define MLVL    4
#define DCH     (DIM / 4)      // 32 k-chunks of 4 for v_wmma_f32_16x16x4_f32
#define TCODES  16             // codes per WMMA tile (N dimension)
#define NTILES  (KCODES / TCODES)  // 64
#define WAVES   8
#define ROWS_PER_BLOCK (WAVES * 16)  // 128
#define CB_STRIDE 132          // padded LDS stride (floats): 132%64==4 -> conflict-free
#define LOSS_SCALE (1.25f / (float)((size_t)N_TOK * DIM))

__global__ void rvq_init_kernel(float* __restrict__ loss) { *loss = 0.0f; }

// ||c||^2 for every code of every level -> d_ws (M*K floats)
__global__ void rvq_e2_kernel(const float* __restrict__ cb, float* __restrict__ e2) {
  int i = blockIdx.x * blockDim.x + threadIdx.x;       // code id over M*K (exact grid)
  const float4* row = (const float4*)(cb + (size_t)i * DIM);
  float s = 0.0f;
  #pragma unroll
  for (int j = 0; j < DIM / 4; ++j) {
    float4 v = row[j];
    s += v.x * v.x + v.y * v.y + v.z * v.z + v.w * v.w;
  }
  e2[i] = s;
}

__global__ __launch_bounds__(256) void rvq_main_kernel(
    const float* __restrict__ x, const float* __restrict__ cb,
    const float* __restrict__ e2, float* __restrict__ y,
    float* __restrict__ loss) {
  __shared__ float s_cb[2][TCODES * CB_STRIDE];  // double-buffered codebook tile
  __shared__ int   s_idx[WAVES][16];             // winning code per row per wave
  __shared__ float s_red[WAVES];                 // loss block reduction

  const int tid  = threadIdx.x;
  const int lane = tid & 31;
  const int wave = tid >> 5;
  const int hgrp = lane >> 4;      // 0 -> k-pair {0,1}; 1 -> k-pair {2,3}
  const int row  = lane & 15;      // this lane's row within the wave's 16-row tile
  const int koff = hgrp * 2;

  const int rowG = blockIdx.x * ROWS_PER_BLOCK + wave * 16 + row;

  // LDS destinations for this thread's two staged float4s (512 float4 per tile)
  const int j1   = tid + 256;
  const int dst0 = (tid >> 5) * CB_STRIDE + ((tid & 31) << 2);
  const int dst1 = (j1 >> 5) * CB_STRIDE + ((j1 & 31) << 2);

  // Residual lives in registers, already in wmma f32 16x16x4 A-fragment layout:
  // afrag[c] holds res[row][4c+koff] (x) and res[row][4c+koff+1] (y).
  v2f afrag[DCH];
  {
    const float* xr = x + (size_t)rowG * DIM + koff;
    #pragma unroll
    for (int c = 0; c < DCH; ++c) afrag[c] = *(const v2f*)(xr + 4 * c);
  }

  float loss_acc = 0.0f;

  for (int lvl = 0; lvl < MLVL; ++lvl) {
    const float* cbL = cb + (size_t)lvl * KCODES * DIM;
    const float* e2L = e2 + lvl * KCODES;

    float best[8];
    int   bidx[8];
    #pragma unroll
    for (int v = 0; v < 8; ++v) { best[v] = -3.4e38f; bidx[v] = 0; }

    // ---- prologue: stage tile 0 into buffer 0 ----
    float4 pre0, pre1;
    {
      const float4* src = (const float4*)cbL;
      pre0 = src[tid];
      pre1 = src[j1];
      *(float4*)(&s_cb[0][dst0]) = pre0;
      *(float4*)(&s_cb[0][dst1]) = pre1;
    }
    __syncthreads();

    for (int t = 0; t < NTILES; ++t) {
      // Issue global loads for tile t+1 now; latency hides behind the WMMAs.
      const bool more = (t + 1) < NTILES;
      if (more) {
        const float4* src = (const float4*)(cbL + (size_t)(t + 1) * TCODES * DIM);
        pre0 = src[tid];
        pre1 = src[j1];
      }

      // ---- 16x16 dot-product tile: 32 accumulating f32 WMMAs over D=128 ----
      v8f acc = {};
      const float* bbase = &s_cb[t & 1][row * CB_STRIDE + koff];  // row == B column
      #pragma unroll
      for (int c = 0; c < DCH; ++c) {
        v2f b = *(const v2f*)(bbase + 4 * c);
        acc = __builtin_amdgcn_wmma_f32_16x16x4_f32(
            false, afrag[c], false, b, (short)0, acc, false, false);
      }

      // score = dot - 0.5*||c||^2 ; argmax == argmin distance. Scan order is
      // increasing code index, so strict '>' keeps the first minimum.
      const int   code = t * TCODES + row;
      const float sc   = -0.5f * e2L[code];
      #pragma unroll
      for (int v = 0; v < 8; ++v) {
        float s = acc[v] + sc;
        if (s > best[v]) { best[v] = s; bidx[v] = code; }
      }

      // Commit the prefetched tile into the other buffer (wait lands here, not
      // before the WMMA block), then one barrier per tile.
      if (more) {
        const int nb = (t + 1) & 1;
        *(float4*)(&s_cb[nb][dst0]) = pre0;
        *(float4*)(&s_cb[nb][dst1]) = pre1;
      }
      __syncthreads();
    }

    // Winner reduction across the 16 lanes (same rows, different codes);
    // masks 1..8 stay inside each half-wave matching the C-layout split.
    #pragma unroll
    for (int v = 0; v < 8; ++v) {
      #pragma unroll
      for (int m = 1; m < 16; m <<= 1) {
        float ob = __shfl_xor(best[v], m, 32);
        int   oi = __shfl_xor(bidx[v], m, 32);
        if (ob > best[v] || (ob == best[v] && oi < bidx[v])) {
          best[v] = ob; bidx[v] = oi;
        }
      }
    }
    if (row == 0) {  // lane 0 publishes rows 0..7, lane 16 publishes rows 8..15
      #pragma unroll
      for (int v = 0; v < 8; ++v) s_idx[wave][hgrp * 8 + v] = bidx[v];
    }
    __syncthreads();

    // Subtract the chosen code; new residual^2 IS this level's loss term
    // (commitment and codebook forward values are both mean((q-r)^2)).
    const int    myIdx = s_idx[wave][row];
    const float* q     = cbL + (size_t)myIdx * DIM + koff;
    #pragma unroll
    for (int c = 0; c < DCH; ++c) {
      v2f qv = *(const v2f*)(q + 4 * c);
      afrag[c].x -= qv.x;
      afrag[c].y -= qv.y;
      loss_acc += afrag[c].x * afrag[c].x + afrag[c].y * afrag[c].y;
    }
  }

  // y = x - final residual  (== sum of selected codes, straight-through forward)
  {
    const float* xr = x + (size_t)rowG * DIM + koff;
    float*       yr = y + (size_t)rowG * DIM + koff;
    #pragma unroll
    for (int c = 0; c < DCH; ++c) {
      v2f xv = *(const v2f*)(xr + 4 * c);
      v2f o;
      o.x = xv.x - afrag[c].x;
      o.y = xv.y - afrag[c].y;
      *(v2f*)(yr + 4 * c) = o;
    }
  }

  // Loss: wave reduce -> block reduce -> one atomicAdd per block
  #pragma unroll
  for (int m = 1; m < 32; m <<= 1) loss_acc += __shfl_xor(loss_acc, m, 32);
  if (lane == 0) s_red[wave] = loss_acc;
  __syncthreads();
  if (tid == 0) {
    float s = 0.0f;
    #pragma unroll
    for (int w = 0; w < WAVES; ++w) s += s_red[w];
    atomicAdd(loss, s * LOSS_SCALE);
  }
}

extern "C" void kernel_launch(void* const* d_in, const int* in_sizes, int n_in,
                              void* d_out, int out_size, void* d_ws, size_t ws_size,
                              hipStream_t stream) {
  (void)in_sizes; (void)n_in; (void)ws_size;
  const float* x  = (const float*)d_in[0];   // [8,4096,128] f32
  const float* cb = (const float*)d_in[1];   // [4,1024,128] f32
  float* y     = (float*)d_out;              // y flat, then scalar loss
  float* lossp = y + (out_size - 1);
  float* e2    = (float*)d_ws;               // M*K floats of scratch

  rvq_init_kernel<<<1, 1, 0, stream>>>(lossp);
  rvq_e2_kernel<<<(MLVL * KCODES) / 256, 256, 0, stream>>>(cb, e2);
  rvq_main_kernel<<<N_TOK / ROWS_PER_BLOCK, 256, 0, stream>>>(x, cb, e2, y, lossp);
}